// StructureExpertGNN_22892175687766
// MI455X (gfx1250) — compile-verified
//
#include <hip/hip_runtime.h>
#include <hip/hip_bf16.h>
#include <math.h>

typedef float v2f __attribute__((ext_vector_type(2)));
typedef float v8f __attribute__((ext_vector_type(8)));

#define GAT_SLOPE 0.2f   // GATv2 attention leaky slope
#define MID_SLOPE 0.01f  // F.leaky_relu default

#define KC 64            // K-chunk staged in LDS per iteration
#define APITCH 66        // LDS row pitch (floats): bank = (2*row + k) % 64, conflict-free

// ---- exact float atomic max via int punning (works for all signs) ----
__device__ __forceinline__ void atomic_max_f32(float* addr, float val) {
  if (val >= 0.0f) {
    atomicMax((int*)addr, __float_as_int(val));
  } else {
    atomicMin((unsigned int*)addr, __float_as_uint(val));
  }
}

// ---------------------------------------------------------------------
// Fill kernel (grid-stride)
// ---------------------------------------------------------------------
__global__ void fill_f32_kernel(float* __restrict__ p, size_t n, float v) {
  size_t i = (size_t)blockIdx.x * blockDim.x + threadIdx.x;
  size_t stride = (size_t)gridDim.x * blockDim.x;
  for (; i < n; i += stride) p[i] = v;
}

// ---------------------------------------------------------------------
// WMMA fp32 GEMM:  C[M,N] = A[M,K] @ W[K,N] + bias[N]
// Block = 128 threads = 4 waves sharing one 16-row M tile.
//  * The block stages A[16 x 64] into LDS with coalesced global_load_b128
//    (4x reuse across the 4 waves), conflict-free pitch-66 layout.
//  * Each wave then feeds V_WMMA_F32_16X16X4_F32 from ds_load_b64 (A) and
//    L2-resident coalesced loads of W (B).
// Requires: M % 16 == 0, N % 64 == 0, K % 64 == 0 (true for all call sites).
// ---------------------------------------------------------------------
__global__ void wmma_gemm_bias_kernel(const float* __restrict__ A,
                                      const float* __restrict__ W,
                                      const float* __restrict__ bias,
                                      float* __restrict__ C,
                                      int M, int N, int K) {
  __shared__ float aTile[16 * APITCH];

  const int tid   = (int)threadIdx.x;   // 0..127
  const int wave  = tid >> 5;           // 0..3
  const int lane  = tid & 31;
  const int nb    = N >> 6;             // number of 64-wide N blocks
  const int tileM = (int)blockIdx.x / nb;
  const int nblk  = (int)blockIdx.x - tileM * nb;
  const int tileN = nblk * 4 + wave;

  const int half  = lane >> 4;          // 0: K pair {0,1}; 1: K pair {2,3}
  const int l16   = lane & 15;
  const int bcol  = tileN * 16 + l16;   // B/C column handled by this lane
  const int rowBase = tileM * 16;

  v8f acc = {};

  for (int kc = 0; kc < K; kc += KC) {
    // ---- cooperative coalesced stage of A[rowBase..+15][kc..kc+63] ----
    for (int i = tid; i < 16 * (KC / 4); i += 128) {  // 256 float4s
      int r  = i >> 4;        // row 0..15
      int k4 = i & 15;        // float4 index within the 64-wide chunk
      float4 v = *(const float4*)(A + (size_t)(rowBase + r) * K + kc + 4 * k4);
      float* dp = &aTile[r * APITCH + 4 * k4];
      dp[0] = v.x; dp[1] = v.y; dp[2] = v.z; dp[3] = v.w;
    }
    // prefetch next A chunk while this one is consumed
    if (kc + KC < K) {
      __builtin_prefetch(A + (size_t)(rowBase + l16) * K + kc + KC, 0, 0);
    }
    __syncthreads();

    // ---- WMMA inner loop: A from LDS, B from (L2-resident) global ----
    const float* __restrict__ arow = &aTile[l16 * APITCH + 2 * half];
    for (int kk = 0; kk < KC; kk += 4) {
      v2f a, b;
      a.x = arow[kk];
      a.y = arow[kk + 1];
      const float* __restrict__ wp = W + (size_t)(kc + kk + 2 * half) * N + bcol;
      b.x = wp[0];
      b.y = wp[N];
      acc = __builtin_amdgcn_wmma_f32_16x16x4_f32(
          /*neg_a=*/false, a, /*neg_b=*/false, b,
          /*c_mod=*/(short)0, acc, /*reuse_a=*/false, /*reuse_b=*/false);
    }
    __syncthreads();
  }

  const float bv    = bias[bcol];
  const int   rbase = rowBase + 8 * half;  // C layout: M = v + 8*(lane/16)
#pragma unroll
  for (int v = 0; v < 8; ++v) {
    C[(size_t)(rbase + v) * N + bcol] = acc[v] + bv;
  }
}

// ---------------------------------------------------------------------
// Edge pass 1: score[e,h] = sum_c att[h,c]*leaky(xl[src,h,c]+xr[dst,h,c],0.2)
// plus segment max into mbuf[dst,h]. One thread per (edge, head).
// Edges e >= E are the implicit self loops (src = dst = e - E).
// ---------------------------------------------------------------------
__global__ void edge_score_max_kernel(const float* __restrict__ xl,
                                      const float* __restrict__ xr,
                                      const int* __restrict__ srcIdx,
                                      const int* __restrict__ dstIdx,
                                      const float* __restrict__ att,
                                      float* __restrict__ score,
                                      float* __restrict__ mbuf,
                                      int E, int Nn, int H, int C) {
  int idx = blockIdx.x * blockDim.x + threadIdx.x;
  int total = (E + Nn) * H;
  if (idx >= total) return;
  int e = idx / H;
  int h = idx - e * H;
  int s = (e < E) ? srcIdx[e] : (e - E);
  int d = (e < E) ? dstIdx[e] : (e - E);

  const float4* pl = (const float4*)(xl + ((size_t)s * H + h) * C);
  const float4* pr = (const float4*)(xr + ((size_t)d * H + h) * C);
  const float4* pa = (const float4*)(att + (size_t)h * C);

  float acc = 0.0f;
  for (int c = 0; c < (C >> 2); ++c) {
    float4 a = pl[c];
    float4 b = pr[c];
    float4 w = pa[c];
    float v0 = a.x + b.x; v0 = (v0 > 0.f) ? v0 : GAT_SLOPE * v0;
    float v1 = a.y + b.y; v1 = (v1 > 0.f) ? v1 : GAT_SLOPE * v1;
    float v2 = a.z + b.z; v2 = (v2 > 0.f) ? v2 : GAT_SLOPE * v2;
    float v3 = a.w + b.w; v3 = (v3 > 0.f) ? v3 : GAT_SLOPE * v3;
    acc += w.x * v0 + w.y * v1 + w.z * v2 + w.w * v3;
  }
  score[idx] = acc;
  atomic_max_f32(&mbuf[(size_t)d * H + h], acc);
}

// ---------------------------------------------------------------------
// Edge pass 2: e = exp(score - m[dst]); segment sum into ssum[dst,h].
// score buffer is overwritten in place with the exponentiated value.
// ---------------------------------------------------------------------
__global__ void edge_exp_sum_kernel(const int* __restrict__ dstIdx,
                                    const float* __restrict__ mbuf,
                                    float* __restrict__ score,
                                    float* __restrict__ ssum,
                                    int E, int Nn, int H) {
  int idx = blockIdx.x * blockDim.x + threadIdx.x;
  int total = (E + Nn) * H;
  if (idx >= total) return;
  int e = idx / H;
  int h = idx - e * H;
  int d = (e < E) ? dstIdx[e] : (e - E);
  float ex = __expf(score[idx] - mbuf[(size_t)d * H + h]);
  score[idx] = ex;
  atomicAdd(&ssum[(size_t)d * H + h], ex);
}

// ---------------------------------------------------------------------
// Edge pass 3: alpha = e / (sum[dst] + 1e-16); out[dst,h,:] += alpha*xl[src,h,:]
// ---------------------------------------------------------------------
__global__ void edge_scatter_kernel(const float* __restrict__ xl,
                                    const int* __restrict__ srcIdx,
                                    const int* __restrict__ dstIdx,
                                    const float* __restrict__ escore,
                                    const float* __restrict__ ssum,
                                    float* __restrict__ outAcc,
                                    int E, int Nn, int H, int C) {
  int idx = blockIdx.x * blockDim.x + threadIdx.x;
  int total = (E + Nn) * H;
  if (idx >= total) return;
  int e = idx / H;
  int h = idx - e * H;
  int s = (e < E) ? srcIdx[e] : (e - E);
  int d = (e < E) ? dstIdx[e] : (e - E);

  float alpha = escore[idx] / (ssum[(size_t)d * H + h] + 1e-16f);
  const float4* pl = (const float4*)(xl + ((size_t)s * H + h) * C);
  float* po = outAcc + ((size_t)d * H + h) * C;
  for (int c = 0; c < (C >> 2); ++c) {
    float4 a = pl[c];
    atomicAdd(po + 4 * c + 0, a.x * alpha);
    atomicAdd(po + 4 * c + 1, a.y * alpha);
    atomicAdd(po + 4 * c + 2, a.z * alpha);
    atomicAdd(po + 4 * c + 3, a.w * alpha);
  }
}

// In-place: p[i] = leaky(p[i] + bias[i % nf], slope)
__global__ void bias_act_kernel(float* __restrict__ p,
                                const float* __restrict__ bias,
                                size_t total, int nf, float slope) {
  size_t i = (size_t)blockIdx.x * blockDim.x + threadIdx.x;
  size_t stride = (size_t)gridDim.x * blockDim.x;
  for (; i < total; i += stride) {
    float v = p[i] + bias[(int)(i % nf)];
    p[i] = (v > 0.f) ? v : slope * v;
  }
}

// outp[i] = in[i] + bias[i % nf]   (no activation)
__global__ void bias_copy_kernel(const float* __restrict__ in,
                                 const float* __restrict__ bias,
                                 float* __restrict__ outp,
                                 size_t total, int nf) {
  size_t i = (size_t)blockIdx.x * blockDim.x + threadIdx.x;
  size_t stride = (size_t)gridDim.x * blockDim.x;
  for (; i < total; i += stride) outp[i] = in[i] + bias[(int)(i % nf)];
}

// ---------------------------------------------------------------------
// Head: logits[n] = (leaky(h[n]@Wp1 + bp1, 0.01)) @ Wp2 + bp2
// One thread per node; Wp1/Wp2 (8 KB) are cache-resident.
// ---------------------------------------------------------------------
__global__ void head_kernel(const float* __restrict__ h,
                            const float* __restrict__ Wp1,
                            const float* __restrict__ bp1,
                            const float* __restrict__ Wp2,
                            const float* __restrict__ bp2,
                            float* __restrict__ logits, int Nn) {
  int n = blockIdx.x * blockDim.x + threadIdx.x;
  if (n >= Nn) return;
  float hv[64];
  const float4* ph = (const float4*)(h + (size_t)n * 64);
#pragma unroll
  for (int c = 0; c < 16; ++c) {
    float4 t = ph[c];
    hv[4 * c + 0] = t.x; hv[4 * c + 1] = t.y;
    hv[4 * c + 2] = t.z; hv[4 * c + 3] = t.w;
  }
  float acc2 = bp2[0];
  for (int j = 0; j < 32; ++j) {
    float z = bp1[j];
#pragma unroll
    for (int c = 0; c < 64; ++c) z += hv[c] * Wp1[c * 32 + j];
    z = (z > 0.f) ? z : MID_SLOPE * z;
    acc2 += z * Wp2[j];
  }
  logits[n] = acc2;
}

// ---------------------------------------------------------------------
extern "C" void kernel_launch(void* const* d_in, const int* in_sizes, int n_in,
                              void* d_out, int out_size, void* d_ws, size_t ws_size,
                              hipStream_t stream) {
  (void)n_in; (void)out_size; (void)ws_size;

  const float* x     = (const float*)d_in[0];
  const int*   ei    = (const int*)d_in[1];
  const float* W1l   = (const float*)d_in[2];
  const float* b1l   = (const float*)d_in[3];
  const float* W1r   = (const float*)d_in[4];
  const float* b1r   = (const float*)d_in[5];
  const float* att1  = (const float*)d_in[6];
  const float* bias1 = (const float*)d_in[7];
  const float* W2l   = (const float*)d_in[8];
  const float* b2l   = (const float*)d_in[9];
  const float* W2r   = (const float*)d_in[10];
  const float* b2r   = (const float*)d_in[11];
  const float* att2  = (const float*)d_in[12];
  const float* bias2 = (const float*)d_in[13];
  const float* Wp1   = (const float*)d_in[14];
  const float* bp1   = (const float*)d_in[15];
  const float* Wp2   = (const float*)d_in[16];
  const float* bp2   = (const float*)d_in[17];

  const int Nn = in_sizes[0] / 256;   // 50000 nodes
  const int E  = in_sizes[1] / 2;     // 600000 edges
  const int ET = E + Nn;              // + self loops

  const int* srcIdx = ei;
  const int* dstIdx = ei + E;

  // ---- workspace layout (floats) ----
  float* ws   = (float*)d_ws;
  size_t NF1  = (size_t)Nn * 256;
  float* xl1  = ws;                  // [N,256]  (layer2: xl2 [N,64] at same base,
  float* xr1  = ws + NF1;            // [N,256]   out2 [N,64] at xl1 + N*64)
  float* out1 = ws + 2 * NF1;        // [N,256]  -> becomes h1 in place
  float* scr  = ws + 3 * NF1;        // [(E+N)*4] edge scores / exp
  float* mbuf = scr + (size_t)ET * 4;  // [N*4]
  float* sbuf = mbuf + (size_t)Nn * 4; // [N*4]
  float* xl2  = xl1;                 // [N,64]
  float* xr2  = xr1;                 // [N,64]
  float* out2 = xl1 + (size_t)Nn * 64; // [N,64]

  float* logits = (float*)d_out;         // [N]
  float* hOut   = (float*)d_out + Nn;    // [N,64]

  const float NEG_INF = -__builtin_inff();
  const int TB = 256;

  // ================= Layer 1 (H=4, C=64, concat) =================
  fill_f32_kernel<<<512, TB, 0, stream>>>(out1, NF1, 0.0f);
  fill_f32_kernel<<<512, TB, 0, stream>>>(mbuf, (size_t)Nn * 4, NEG_INF);
  fill_f32_kernel<<<512, TB, 0, stream>>>(sbuf, (size_t)Nn * 4, 0.0f);

  {
    int blocks = ((Nn + 15) / 16) * (256 / 64);   // one block per 16x64 strip
    wmma_gemm_bias_kernel<<<blocks, 128, 0, stream>>>(x, W1l, b1l, xl1, Nn, 256, 256);
    wmma_gemm_bias_kernel<<<blocks, 128, 0, stream>>>(x, W1r, b1r, xr1, Nn, 256, 256);
  }
  {
    int total = ET * 4;
    int blocks = (total + TB - 1) / TB;
    edge_score_max_kernel<<<blocks, TB, 0, stream>>>(xl1, xr1, srcIdx, dstIdx, att1,
                                                     scr, mbuf, E, Nn, 4, 64);
    edge_exp_sum_kernel<<<blocks, TB, 0, stream>>>(dstIdx, mbuf, scr, sbuf, E, Nn, 4);
    edge_scatter_kernel<<<blocks, TB, 0, stream>>>(xl1, srcIdx, dstIdx, scr, sbuf,
                                                   out1, E, Nn, 4, 64);
  }
  // h1 = leaky(out1 + bias1, 0.01), in place
  bias_act_kernel<<<512, TB, 0, stream>>>(out1, bias1, NF1, 256, MID_SLOPE);
  float* h1 = out1;

  // ================= Layer 2 (H=1, C=64, mean==identity) =================
  fill_f32_kernel<<<512, TB, 0, stream>>>(out2, (size_t)Nn * 64, 0.0f);
  fill_f32_kernel<<<512, TB, 0, stream>>>(mbuf, (size_t)Nn, NEG_INF);
  fill_f32_kernel<<<512, TB, 0, stream>>>(sbuf, (size_t)Nn, 0.0f);

  {
    int blocks = ((Nn + 15) / 16) * (64 / 64);
    wmma_gemm_bias_kernel<<<blocks, 128, 0, stream>>>(h1, W2l, b2l, xl2, Nn, 64, 256);
    wmma_gemm_bias_kernel<<<blocks, 128, 0, stream>>>(h1, W2r, b2r, xr2, Nn, 64, 256);
  }
  {
    int total = ET;
    int blocks = (total + TB - 1) / TB;
    edge_score_max_kernel<<<blocks, TB, 0, stream>>>(xl2, xr2, srcIdx, dstIdx, att2,
                                                     scr, mbuf, E, Nn, 1, 64);
    edge_exp_sum_kernel<<<blocks, TB, 0, stream>>>(dstIdx, mbuf, scr, sbuf, E, Nn, 1);
    edge_scatter_kernel<<<blocks, TB, 0, stream>>>(xl2, srcIdx, dstIdx, scr, sbuf,
                                                   out2, E, Nn, 1, 64);
  }
  // h = out2 + bias2  -> second output
  bias_copy_kernel<<<512, TB, 0, stream>>>(out2, bias2, hOut, (size_t)Nn * 64, 64);

  // head -> logits (first output)
  head_kernel<<<(Nn + TB - 1) / TB, TB, 0, stream>>>(hOut, Wp1, bp1, Wp2, bp2, logits, Nn);
}